// MB_projection_18708877541408
// MI455X (gfx1250) — compile-verified
//
#include <hip/hip_runtime.h>

typedef __attribute__((ext_vector_type(2))) float v2f;
typedef __attribute__((ext_vector_type(8))) float v8f;
typedef __attribute__((ext_vector_type(4))) int   v4i;

#if __has_builtin(__builtin_amdgcn_global_load_async_to_lds_b128)
#define HAVE_ASYNC_LDS 1
#else
#define HAVE_ASYNC_LDS 0
#endif

#define BATCH 4096
#define FEAT  512
#define OUT   10240
#define NNZ   32
#define TOPK  32

#define CHUNK 32          // K-chunk densified per step
#define APAD  516         // FEAT + 4  -> distinct banks across the 16 A-rows
#define WPAD  36          // CHUNK + 4 -> distinct banks across the 16 B-cols
#define NWAVE 8           // waves per block (wave32)

// ---------------------------------------------------------------------------
// Kernel 1: densify sparse W tiles into LDS and run f32 WMMA GEMM
//   out[b, o] = sum_k input[b, k] * Wd[o, k]
// block = 256 thr (8 waves); block covers 16 batch rows x 128 output cols
// wave  = one 16x16 output tile, K-loop over 512 in chunks of 32 (8 wmma/chunk)
// No inner barriers: wd is wave-private and the LDS pipe is in-order per wave.
// ---------------------------------------------------------------------------
__global__ __launch_bounds__(256) void proj_wmma_kernel(
    const float* __restrict__ input,
    const float* __restrict__ wvals,
    const int*   __restrict__ widx,
    float* __restrict__ out)
{
  __shared__ float a_sh[16 * APAD];            // input tile  16 x 512 (padded)
  __shared__ float w_sh[NWAVE * 16 * WPAD];    // per-wave densified W chunk

  const int tid  = threadIdx.x;
  const int lane = tid & 31;
  const int wave = tid >> 5;
  const int row0 = blockIdx.x * 16;                       // batch tile base
  const int col0 = (blockIdx.y * NWAVE + wave) * 16;      // output-col tile base

  // ---- A tile (16 x 512 f32) HBM -> LDS ----
  for (int i = tid; i < 16 * (FEAT / 4); i += 256) {
    const int r  = i >> 7;           // FEAT/4 == 128
    const int c4 = i & 127;
    const float* src = input + (size_t)(row0 + r) * FEAT + c4 * 4;
    float* dst = &a_sh[r * APAD + c4 * 4];
#if HAVE_ASYNC_LDS
    __builtin_amdgcn_global_load_async_to_lds_b128(
        (__attribute__((address_space(1))) v4i*)(void*)src,
        (__attribute__((address_space(3))) v4i*)(void*)dst, 0, 0);
#else
    *(float4*)dst = *(const float4*)src;
#endif
  }

  // ---- preload this wave's 16 columns of COO weights into registers ----
  // lane L handles column (L>>1), entries p = (L&1)*16 .. +15
  const int    mycol = col0 + (lane >> 1);
  const int    p0    = (lane & 1) * 16;
  const size_t wbase = (size_t)mycol * NNZ + p0;
  __builtin_prefetch(widx + wbase, 0, 0);      // global_prefetch_b8
  float vv[16];
  int   iv[16];
#pragma unroll
  for (int p = 0; p < 16; ++p) {
    vv[p] = wvals[wbase + p];
    iv[p] = widx [wbase + p];
  }

#if HAVE_ASYNC_LDS
#if __has_builtin(__builtin_amdgcn_s_wait_asynccnt)
  __builtin_amdgcn_s_wait_asynccnt(0);
#else
  asm volatile("s_wait_asynccnt 0x0" ::: "memory");
#endif
#endif
  __syncthreads();                             // a_sh visible to all waves

  float*    wd    = &w_sh[wave * 16 * WPAD];
  const int arow  = lane & 15;                 // A fragment: M = lane%16
  const int khalf = (lane >> 4) * 2;           // upper half-wave holds K+2,K+3
  const int bcol  = lane & 15;                 // B fragment: N = lane%16
  const int scol  = lane >> 1;                 // scatter column for this lane

  v8f acc = {};

  for (int k0 = 0; k0 < FEAT; k0 += CHUNK) {
    // zero my contiguous 16-float strip (lane pair covers all 32 k of a col):
    // 4 x ds_store_b128, 16B aligned
    float4* wz = (float4*)&wd[scol * WPAD + p0];
#pragma unroll
    for (int t = 0; t < 4; ++t) wz[t] = float4{0.f, 0.f, 0.f, 0.f};

    // scatter-add entries landing in [k0, k0+CHUNK); ds_add_f32 handles
    // COO duplicates and the lane-pair overlap
#pragma unroll
    for (int p = 0; p < 16; ++p) {
      const int off = iv[p] - k0;
      if ((unsigned)off < (unsigned)CHUNK)
        atomicAdd(&wd[scol * WPAD + off], vv[p]);
    }
    asm volatile("" ::: "memory");   // keep LDS program order (in-order per wave)

    // 8 x V_WMMA_F32_16X16X4_F32 (exact f32 accumulation)
#pragma unroll
    for (int kk = 0; kk < CHUNK; kk += 4) {
      v2f a, b;
      const float* ap = &a_sh[arow * APAD + k0 + kk + khalf];
      a.x = ap[0]; a.y = ap[1];
      const float* bp = &wd[bcol * WPAD + kk + khalf];
      b.x = bp[0]; b.y = bp[1];
      acc = __builtin_amdgcn_wmma_f32_16x16x4_f32(
          false, a, false, b, (short)0, acc, false, false);
    }
    asm volatile("" ::: "memory");
  }

  // ---- store C tile: VGPR r -> M = r + 8*(lane>=16), N = lane%16 ----
  const int n     = lane & 15;
  const int mbase = (lane >> 4) * 8;
#pragma unroll
  for (int r = 0; r < 8; ++r)
    out[(size_t)(row0 + mbase + r) * OUT + col0 + n] = acc[r];
}

// ---------------------------------------------------------------------------
// Kernel 2: per-row top-K filter. One block per batch row; each thread keeps
// 40 elements in registers (as order-preserving sortable uints), block
// binary-searches the 32nd-largest value, then rewrites the row in place.
// One barrier per search iteration (per-iteration counters, pre-zeroed).
// ---------------------------------------------------------------------------
#define EPT (OUT / 256)          // 40 elements per thread
#define FPT (OUT / (256 * 4))    // 10 float4 per thread

__global__ __launch_bounds__(256) void wta_filter_kernel(float* __restrict__ out)
{
  __shared__ unsigned cntArr[32];
  const int tid = threadIdx.x;
  float* rp = out + (size_t)blockIdx.x * OUT;

  if (tid < 32) cntArr[tid] = 0u;

  float    x[EPT];
  unsigned u[EPT];
#pragma unroll
  for (int j = 0; j < FPT; ++j) {
    const float4 v = ((const float4*)rp)[tid + j * 256];
    x[j * 4 + 0] = v.x; x[j * 4 + 1] = v.y;
    x[j * 4 + 2] = v.z; x[j * 4 + 3] = v.w;
  }
#pragma unroll
  for (int e = 0; e < EPT; ++e) {
    const unsigned b = __float_as_uint(x[e]);
    u[e] = b ^ ((0u - (b >> 31)) | 0x80000000u);   // monotonic float->uint
  }
  __syncthreads();

  // binary search: largest t with count(u >= t) >= TOPK
  unsigned lo = 0u, hi = 0xFFFFFFFFu;
  for (int it = 0; it < 32; ++it) {
    const unsigned long long span = (unsigned long long)hi - lo + 1ull;
    const unsigned mid = lo + (unsigned)(span >> 1);
    unsigned local = 0;
#pragma unroll
    for (int e = 0; e < EPT; ++e) local += (u[e] >= mid) ? 1u : 0u;
    atomicAdd(&cntArr[it], local);
    __syncthreads();
    const unsigned c = cntArr[it];     // next iter uses a fresh counter
    if (c >= TOPK) lo = mid; else hi = mid - 1u;
  }
  const unsigned thr = lo;

  // keep top-K (values >= threshold), zero the rest
#pragma unroll
  for (int j = 0; j < FPT; ++j) {
    float4 v;
    v.x = (u[j * 4 + 0] >= thr) ? x[j * 4 + 0] : 0.0f;
    v.y = (u[j * 4 + 1] >= thr) ? x[j * 4 + 1] : 0.0f;
    v.z = (u[j * 4 + 2] >= thr) ? x[j * 4 + 2] : 0.0f;
    v.w = (u[j * 4 + 3] >= thr) ? x[j * 4 + 3] : 0.0f;
    ((float4*)rp)[tid + j * 256] = v;
  }
}

// ---------------------------------------------------------------------------
extern "C" void kernel_launch(void* const* d_in, const int* in_sizes, int n_in,
                              void* d_out, int out_size, void* d_ws, size_t ws_size,
                              hipStream_t stream) {
  (void)in_sizes; (void)n_in; (void)out_size; (void)d_ws; (void)ws_size;
  const float* input = (const float*)d_in[0];
  const float* wvals = (const float*)d_in[1];
  const int*   widx  = (const int*)  d_in[2];
  float* out = (float*)d_out;

  dim3 g1(BATCH / 16, OUT / (16 * NWAVE));   // 256 x 80 blocks
  proj_wmma_kernel<<<g1, 256, 0, stream>>>(input, wvals, widx, out);
  wta_filter_kernel<<<BATCH, 256, 0, stream>>>(out);
}